// CorefGRU_39496519254162
// MI455X (gfx1250) — compile-verified
//
#include <hip/hip_runtime.h>
#include <hip/hip_bf16.h>
#include <math.h>

#define B_    8
#define N_    512
#define DIN_  256
#define R_    2
#define RDIMS_ 64
#define C_    128
#define DOUT_ 128      // R*RDIMS
#define G3_   384      // 3*DOUT
#define BN_   (B_ * N_)   // 4096

typedef float v2f __attribute__((ext_vector_type(2)));
typedef float v8f __attribute__((ext_vector_type(8)));

// ---------------------------------------------------------------------------
// Kernel A: Xp = X @ Wstacked  via V_WMMA_F32_16X16X4_F32 (f32 precision kept)
// X: (4096, 256) row-major, W: (256, 384) row-major, Xp: (4096, 384)
// One wave computes a 16x64 strip of Xp = FOUR independent 16x16 accumulator
// tiles -> 4 parallel WMMA dependency chains (hides XDL + load latency) and
// 4x reuse of each A fragment.
// A frag (16x4 f32, 2 VGPRs): lanes 0-15 -> M=lane,    K=k0+{0,1};
//                             lanes 16-31 -> M=lane-16, K=k0+{2,3}.
// B frag (4x16 f32, 2 VGPRs): lanes 0-15 -> N=lane,    K=k0+{0,1};
//                             lanes 16-31 -> N=lane-16, K=k0+{2,3}.
// D (16x16 f32, 8 VGPRs): lane<16 -> N=lane, rows M=0..7;
//                         lane>=16 -> N=lane-16, rows M=8..15.
// ---------------------------------------------------------------------------
__global__ __launch_bounds__(128) void xp_wmma_kernel(
    const float* __restrict__ X, const float* __restrict__ W,
    float* __restrict__ Xp)
{
  const int lane = threadIdx.x & 31;
  const int wave = threadIdx.x >> 5;
  const int task = blockIdx.x * 4 + wave;          // 1536 wave-tasks
  const int NG   = G3_ / 64;                       // 6 N-groups of 64 cols
  const int tm   = task / NG;
  const int ng   = task % NG;
  const int sub  = lane & 15;
  const int hi   = lane >> 4;                      // 0 or 1
  const int row  = tm * 16 + sub;                  // A-matrix row (M)
  const int col0 = ng * 64 + sub;                  // first of 4 D columns
  const int khalf = hi * 2;

  const float* __restrict__ xrow = X + (size_t)row * DIN_;
  v8f acc0 = {}, acc1 = {}, acc2 = {}, acc3 = {};

  #pragma unroll 4
  for (int k0 = 0; k0 < DIN_; k0 += 4) {
    const int k = k0 + khalf;
    v2f a = *(const v2f*)(xrow + k);               // contiguous, 8B aligned
    const float* __restrict__ wr0 = W + (size_t)k * G3_ + col0;
    const float* __restrict__ wr1 = wr0 + G3_;
    v2f b0 = { wr0[ 0], wr1[ 0] };
    v2f b1 = { wr0[16], wr1[16] };
    v2f b2 = { wr0[32], wr1[32] };
    v2f b3 = { wr0[48], wr1[48] };
    if (k0 + 8 < DIN_)                             // prefetch next B rows
      __builtin_prefetch(wr0 + (size_t)8 * G3_, 0, 1);
    acc0 = __builtin_amdgcn_wmma_f32_16x16x4_f32(
        false, a, false, b0, (short)0, acc0, false, false);
    acc1 = __builtin_amdgcn_wmma_f32_16x16x4_f32(
        false, a, false, b1, (short)0, acc1, false, false);
    acc2 = __builtin_amdgcn_wmma_f32_16x16x4_f32(
        false, a, false, b2, (short)0, acc2, false, false);
    acc3 = __builtin_amdgcn_wmma_f32_16x16x4_f32(
        false, a, false, b3, (short)0, acc3, false, false);
  }

  const int mbase = tm * 16 + hi * 8;
  float* __restrict__ dst = Xp + (size_t)mbase * G3_ + col0;
  #pragma unroll
  for (int vi = 0; vi < 8; ++vi) {
    dst[(size_t)vi * G3_ +  0] = acc0[vi];
    dst[(size_t)vi * G3_ + 16] = acc1[vi];
    dst[(size_t)vi * G3_ + 32] = acc2[vi];
    dst[(size_t)vi * G3_ + 48] = acc3[vi];
  }
}

// ---------------------------------------------------------------------------
// Kernel B: xatt[i][r] = X[i,:] . Watt[r,:]   (4096 x 2) — one wave per row
// ---------------------------------------------------------------------------
__global__ __launch_bounds__(128) void xatt_kernel(
    const float* __restrict__ X, const float* __restrict__ Watt,
    float* __restrict__ xatt)
{
  const int lane = threadIdx.x & 31;
  const int wave = threadIdx.x >> 5;
  const int i = blockIdx.x * 4 + wave;
  const float* __restrict__ xrow = X + (size_t)i * DIN_;
  float a0 = 0.f, a1 = 0.f;
  #pragma unroll 4
  for (int k = lane; k < DIN_; k += 32) {
    const float xv = xrow[k];
    a0 = fmaf(xv, Watt[k], a0);
    a1 = fmaf(xv, Watt[DIN_ + k], a1);
  }
  #pragma unroll
  for (int off = 16; off; off >>= 1) {
    a0 += __shfl_xor(a0, off, 32);
    a1 += __shfl_xor(a1, off, 32);
  }
  if (lane == 0) {
    xatt[(size_t)i * 2 + 0] = a0;
    xatt[(size_t)i * 2 + 1] = a1;
  }
}

// ---------------------------------------------------------------------------
// Kernel C: sequential scan. One workgroup per batch b (grid = 8), 256 thr.
// Entire recurrent memory state (128x64 f32 = 32KB) lives in LDS.
// ---------------------------------------------------------------------------
__global__ __launch_bounds__(256) void scan_kernel(
    const float* __restrict__ Mmask,   // (B,N)
    const float* __restrict__ Ei,      // (B,N,C)
    const float* __restrict__ Eo,      // (B,N,C)
    const int*   __restrict__ Ri,      // (B,N,C)
    const int*   __restrict__ Ro,      // (B,N,C)
    const float* __restrict__ U,       // (128,384)
    const float* __restrict__ bvec,    // (128)
    const float* __restrict__ Xp,      // (B*N,384) workspace
    const float* __restrict__ xatt,    // (B*N,2)  workspace
    float* __restrict__ outs,          // (B,N,128)
    float* __restrict__ mems,          // (B,N,128,64)
    float* __restrict__ aggs)          // (B,N,2)
{
  __shared__ float m_s[C_ * RDIMS_];   // 32 KB memory state m[c][d]
  __shared__ float h_s[DOUT_];
  __shared__ float prev_s[DOUT_];
  __shared__ float xp_s[G3_];
  __shared__ float hh_s[G3_];
  __shared__ float alpha_s[C_];
  __shared__ float eo_s[C_];
  __shared__ int   ri_s[C_];
  __shared__ int   ro_s[C_];
  __shared__ float red_s[16];

  const int b = blockIdx.x;
  const int t = threadIdx.x;

  for (int i = t; i < C_ * RDIMS_; i += 256) m_s[i] = 0.f;
  if (t < DOUT_) h_s[t] = 0.f;
  __syncthreads();

  for (int n = 0; n < N_; ++n) {
    const size_t bn = (size_t)b * N_ + n;
    const float mval = Mmask[bn];

    // ---- stage per-step rows ----------------------------------------------
    float am = 0.f;
    if (t < C_) {
      ri_s[t] = Ri[bn * C_ + t];
      ro_s[t] = Ro[bn * C_ + t];
      eo_s[t] = Eo[bn * C_ + t];
      const float actv = xatt[bn * 2 + ri_s[t]];   // one-hot @ Watt @ x
      am = __expf(actv) * Ei[bn * C_ + t] + (float)1e-100;  // EPS flushes, as in f32 jnp
    }
    for (int j = t; j < G3_; j += 256) xp_s[j] = Xp[bn * G3_ + j];

    // ---- softmax denominator (wave32 shuffle reduce, 4 partials) ----------
    float wsum = am;
    #pragma unroll
    for (int off = 16; off; off >>= 1) wsum += __shfl_xor(wsum, off, 32);
    if (t < C_ && (t & 31) == 0) red_s[t >> 5] = wsum;
    __syncthreads();                                              // B1

    const float denom = red_s[0] + red_s[1] + red_s[2] + red_s[3];
    const float alpha = (t < C_) ? am / denom : 0.f;
    if (t < C_) alpha_s[t] = alpha;

    // ---- agg[r] = sum_{c: ri==r} alpha[c] ---------------------------------
    float g0 = (t < C_ && ri_s[t] == 0) ? alpha : 0.f;
    float g1 = (t < C_ && ri_s[t] == 1) ? alpha : 0.f;
    #pragma unroll
    for (int off = 16; off; off >>= 1) {
      g0 += __shfl_xor(g0, off, 32);
      g1 += __shfl_xor(g1, off, 32);
    }
    if (t < C_ && (t & 31) == 0) {
      red_s[4 + (t >> 5)] = g0;
      red_s[8 + (t >> 5)] = g1;
    }
    __syncthreads();                                              // B2
    if (t == 0) aggs[bn * 2 + 0] = red_s[4] + red_s[5] + red_s[6] + red_s[7];
    if (t == 1) aggs[bn * 2 + 1] = red_s[8] + red_s[9] + red_s[10] + red_s[11];

    // ---- prev = aggw @ mprev : prev[r*64+d] = sum_c aggw[r][c]*m[c][d] ----
    if (t < DOUT_) {
      const int r = t >> 6, d = t & 63;
      float acc = 0.f;
      #pragma unroll 4
      for (int c = 0; c < C_; ++c) {
        const float w = (ri_s[c] == r) ? alpha_s[c] : 0.f;
        acc = fmaf(w, m_s[c * RDIMS_ + d], acc);   // bank-conflict-free (stride 64)
      }
      prev_s[t] = acc;
    }
    __syncthreads();                                              // B3

    // ---- hh = prev @ U (1x128 @ 128x384); U resident in L2 ----------------
    for (int j = t; j < G3_; j += 256) {
      float acc = 0.f;
      #pragma unroll 8
      for (int k = 0; k < DOUT_; ++k)
        acc = fmaf(prev_s[k], U[(size_t)k * G3_ + j], acc);
      hh_s[j] = acc;
    }
    __syncthreads();                                              // B4

    // ---- GRU gates --------------------------------------------------------
    if (t < DOUT_) {
      const float bb  = bvec[t];
      const float r_g = 1.f / (1.f + __expf(-(xp_s[t] + hh_s[t] + bb)));
      const float z_g = 1.f / (1.f + __expf(-(xp_s[DOUT_ + t] + hh_s[DOUT_ + t] + bb)));
      const float h_t = tanhf(xp_s[2 * DOUT_ + t] + r_g * hh_s[2 * DOUT_ + t] + bb);
      float hn = z_g * prev_s[t] + (1.f - z_g) * h_t;
      hn = (1.f - mval) * h_s[t] + mval * hn;      // mask blend
      h_s[t] = hn;
      outs[bn * DOUT_ + t] = hn;
    }
    __syncthreads();                                              // B5

    // ---- mnew[c][d] = (1-e)*m[c][d] + e*hnew_r[Ro[c]][d]; write mems ------
    for (int i = t; i < C_ * RDIMS_; i += 256) {
      const int c = i >> 6;
      const int d = i & 63;
      const float e  = mval * eo_s[c];
      const float hv = h_s[ro_s[c] * RDIMS_ + d];
      const float mv = (1.f - e) * m_s[i] + e * hv;
      m_s[i] = mv;
      mems[bn * (size_t)(C_ * RDIMS_) + i] = mv;
    }
    __syncthreads();                                              // B6
  }
}

// ---------------------------------------------------------------------------
extern "C" void kernel_launch(void* const* d_in, const int* in_sizes, int n_in,
                              void* d_out, int out_size, void* d_ws, size_t ws_size,
                              hipStream_t stream) {
  (void)in_sizes; (void)n_in; (void)out_size; (void)ws_size;
  const float* X     = (const float*)d_in[0];
  const float* Mmask = (const float*)d_in[1];
  const float* Ei    = (const float*)d_in[2];
  const float* Eo    = (const float*)d_in[3];
  const int*   Ri    = (const int*)  d_in[4];
  const int*   Ro    = (const int*)  d_in[5];
  const float* W     = (const float*)d_in[6];
  const float* U     = (const float*)d_in[7];
  const float* bvec  = (const float*)d_in[8];
  const float* Watt  = (const float*)d_in[9];

  float* xp_ws   = (float*)d_ws;                       // 4096*384 f32
  float* xatt_ws = xp_ws + (size_t)BN_ * G3_;          // 4096*2 f32

  float* outs = (float*)d_out;                                   // 524288
  float* mems = outs + (size_t)B_ * N_ * DOUT_;                  // 33554432
  float* aggs = mems + (size_t)B_ * N_ * C_ * RDIMS_;            // 8192

  // 1536 wave-tasks (16x64 strips) / 4 waves per block
  xp_wmma_kernel<<<(BN_ / 16) * (G3_ / 64) / 4, 128, 0, stream>>>(X, W, xp_ws);
  xatt_kernel<<<BN_ / 4, 128, 0, stream>>>(X, Watt, xatt_ws);
  scan_kernel<<<B_, 256, 0, stream>>>(Mmask, Ei, Eo, Ri, Ro, U, bvec,
                                      xp_ws, xatt_ws, outs, mems, aggs);
}